// GCNLayer_62423054680357
// MI455X (gfx1250) — compile-verified
//
#include <hip/hip_runtime.h>
#include <hip/hip_bf16.h>

typedef __attribute__((ext_vector_type(2))) float v2f;
typedef __attribute__((ext_vector_type(8))) float v8f;

#define DIM 256
#define LDS_STRIDE 260   // 256 + 4 pad: lane L column reads hit bank (4L+k)%64 -> conflict-free

// ---------------------------------------------------------------------------
// Phase 1: S = X @ W   (fp32 GEMM via V_WMMA_F32_16X16X4_F32)
// block = 512 threads = 16 waves. Block b computes rows [16b, 16b+16).
// Wave w computes the 16x16 tile at columns [16w, 16w+16).
// ---------------------------------------------------------------------------
__global__ __launch_bounds__(512) void gcn_gemm_xw(const float* __restrict__ X,
                                                   const float* __restrict__ W,
                                                   float* __restrict__ S) {
    const int m0 = blockIdx.x * 16;
    __shared__ float Atile[16 * LDS_STRIDE];

    // Cooperative load of 16x256 fp32 X stripe: 1024 float4 loads, 2 per thread.
    {
        const int t = threadIdx.x;
        #pragma unroll
        for (int i = 0; i < 2; ++i) {
            const int v   = t + i * 512;     // vec4 index 0..1023
            const int row = v >> 6;          // 64 float4 per row
            const int c4  = (v & 63) * 4;
            const float4 x =
                ((const float4*)(X + (size_t)(m0 + row) * DIM))[v & 63];
            float* dst = &Atile[row * LDS_STRIDE + c4];
            dst[0] = x.x; dst[1] = x.y; dst[2] = x.z; dst[3] = x.w;
        }
    }
    __syncthreads();

    const int wave = threadIdx.x >> 5;   // 0..15
    const int lane = threadIdx.x & 31;
    const int n0   = wave * 16;
    const int col  = lane & 15;          // N (and A-row) index within tile
    const int koff = (lane >> 4) * 2;    // half-wave K split: {0,1} / {2,3}

    v8f acc = {};
    const float* wcol = W + n0 + col;    // &W[k*256 + n0+col]
    const float* arow = &Atile[col * LDS_STRIDE];

    #pragma unroll 4
    for (int k0 = 0; k0 < DIM; k0 += 4) {
        const int k = k0 + koff;
        v2f a, bm;
        // A 16x4 fp32 layout: VGPR0 = K=koff, VGPR1 = K=koff+1 (per half-wave)
        a.x = arow[k];
        a.y = arow[k + 1];
        // B 4x16 fp32 layout (mirrored): rows K=koff, K=koff+1; N = lane&15
        bm.x = wcol[(size_t)k * DIM];
        bm.y = wcol[(size_t)(k + 1) * DIM];
        acc = __builtin_amdgcn_wmma_f32_16x16x4_f32(
            /*neg_a=*/false, a, /*neg_b=*/false, bm,
            /*c_mod=*/(short)0, acc, /*reuse_a=*/false, /*reuse_b=*/false);
    }

    // C/D layout: VGPR r -> M = r + 8*(lane>=16), N = lane&15
    const int mbase = m0 + ((lane >> 4) << 3);
    float* outp = S + (size_t)mbase * DIM + n0 + col;
    #pragma unroll
    for (int r = 0; r < 8; ++r) {
        outp[(size_t)r * DIM] = acc[r];
    }
}

// ---------------------------------------------------------------------------
// Phase 2a: out[i, :] = b[:]   (seed accumulator with bias)
// ---------------------------------------------------------------------------
__global__ __launch_bounds__(256) void gcn_init_out(float* __restrict__ out,
                                                    const float* __restrict__ b,
                                                    long long total) {
    long long i = (long long)blockIdx.x * blockDim.x + threadIdx.x;
    if (i < total) out[i] = b[i & (DIM - 1)];
}

// ---------------------------------------------------------------------------
// Native no-return fp32 atomic add (GLOBAL_ATOMIC_ADD_F32, STOREcnt-tracked).
// Inline asm guarantees we never fall back to a CAS retry loop.
// ---------------------------------------------------------------------------
__device__ __forceinline__ void atomic_add_f32_native(float* addr, float val) {
    asm volatile("global_atomic_add_f32 %0, %1, off"
                 :
                 : "v"(addr), "v"(val)
                 : "memory");
}

// ---------------------------------------------------------------------------
// Phase 2b: for each edge e: out[dst[e], :] += S[src[e], :] * val[e]
// One wave32 per edge; each lane owns 8 consecutive floats of the 256-row.
// ---------------------------------------------------------------------------
__global__ __launch_bounds__(256) void gcn_scatter(const float* __restrict__ S,
                                                   const int* __restrict__ esrc,
                                                   const int* __restrict__ edst,
                                                   const float* __restrict__ eval,
                                                   float* __restrict__ out,
                                                   int n_edges) {
    const int gwave = (int)(((long long)blockIdx.x * blockDim.x + threadIdx.x) >> 5);
    const int lane  = threadIdx.x & 31;
    if (gwave >= n_edges) return;

    // Wave-uniform edge id -> scalar loads for src/dst/val
    const int e = __builtin_amdgcn_readfirstlane(gwave);
    const int s = esrc[e];
    const int d = edst[e];
    const float v = eval[e];

    const float4* sp = (const float4*)(S + (size_t)s * DIM) + lane * 2;
    const float4 a = sp[0];
    const float4 c = sp[1];

    float* op = out + (size_t)d * DIM + lane * 8;
    atomic_add_f32_native(op + 0, a.x * v);
    atomic_add_f32_native(op + 1, a.y * v);
    atomic_add_f32_native(op + 2, a.z * v);
    atomic_add_f32_native(op + 3, a.w * v);
    atomic_add_f32_native(op + 4, c.x * v);
    atomic_add_f32_native(op + 5, c.y * v);
    atomic_add_f32_native(op + 6, c.z * v);
    atomic_add_f32_native(op + 7, c.w * v);
}

// ---------------------------------------------------------------------------
extern "C" void kernel_launch(void* const* d_in, const int* in_sizes, int n_in,
                              void* d_out, int out_size, void* d_ws, size_t ws_size,
                              hipStream_t stream) {
    const float* X    = (const float*)d_in[0];  // [N, 256]
    const int*   esrc = (const int*)d_in[1];    // [E]
    const int*   edst = (const int*)d_in[2];    // [E]
    const float* eval = (const float*)d_in[3];  // [E]
    const float* W    = (const float*)d_in[4];  // [256, 256]
    const float* b    = (const float*)d_in[5];  // [256]
    float* out = (float*)d_out;                 // [N, 256]
    float* S   = (float*)d_ws;                  // scratch: [N, 256] fp32 = 102.4 MB

    const int n_nodes = in_sizes[0] / DIM;      // 100000
    const int n_edges = in_sizes[1];            // 3200000

    // Phase 1: S = X @ W. 16 rows per block; 100000/16 = 6250 exactly.
    gcn_gemm_xw<<<n_nodes / 16, 512, 0, stream>>>(X, W, S);

    // Phase 2a: out = broadcast(b)
    const long long total = (long long)n_nodes * DIM;
    const int init_blocks = (int)((total + 255) / 256);
    gcn_init_out<<<init_blocks, 256, 0, stream>>>(out, b, total);

    // Phase 2b: scatter-add, one wave per edge (8 waves per 256-thread block)
    const int waves_per_block = 256 / 32;
    const int sc_blocks = (n_edges + waves_per_block - 1) / waves_per_block;
    gcn_scatter<<<sc_blocks, 256, 0, stream>>>(S, esrc, edst, eval, out, n_edges);
}